// GraphNorm_5016521802061
// MI455X (gfx1250) — compile-verified
//
#include <hip/hip_runtime.h>
#include <hip/hip_bf16.h>
#include <cstdint>

// Problem constants (from reference)
#define N_ROWS   1048576
#define DIM      128
#define NSEG     512
#define EPS_F    1e-6f

// Pass-1/3 tiling: 1 wave = 1 row (32 lanes x float4 = 128 floats)
#define TILE_ROWS 8               // rows per async tile (8 x 512B = 4KB)
#define RPW       128             // rows per wave
#define WAVES     8               // waves per block (256 threads, wave32)
#define TILES     (RPW / TILE_ROWS)      // 16
#define ROWS_PER_BLOCK (RPW * WAVES)     // 1024

// Native clang vector type (required by the nontemporal builtins; same 16B
// layout as float4)
typedef float v4f __attribute__((ext_vector_type(4)));

// ---------------------------------------------------------------------------
// CDNA5 async global->LDS copy (gfx1250). One instruction moves 32 lanes x
// 16B = 512B = one row. Inline asm per bridge doc (portable across
// toolchains; bypasses builtin-arity differences).
// Generic pointers to __shared__ carry the LDS byte offset in their low 32
// bits (shared aperture base occupies addr[63:32]), so the truncated value
// is exactly the VDST LDS address the instruction wants.
// ---------------------------------------------------------------------------
__device__ __forceinline__ void async_load_row_b128(const void* gsrc, void* ldst) {
  uint32_t loff = (uint32_t)(uintptr_t)ldst;           // LDS byte offset
  uint64_t gaddr = (uint64_t)(uintptr_t)gsrc;          // 64-bit global addr
  asm volatile("global_load_async_to_lds_b128 %0, %1, off"
               :: "v"(loff), "v"(gaddr)
               : "memory");
}

__device__ __forceinline__ void wait_async_le8() {
  // <=8 outstanding: async loads complete in order, so the older 8-op tile is done
  asm volatile("s_wait_asynccnt 8" ::: "memory");
}
__device__ __forceinline__ void wait_async_le0() {
  asm volatile("s_wait_asynccnt 0" ::: "memory");
}

__device__ __forceinline__ void atom_add_f32(float* p, float v) {
  // relaxed device-scope fadd -> native global_atomic_add_f32 on gfx1250
  __hip_atomic_fetch_add(p, v, __ATOMIC_RELAXED, __HIP_MEMORY_SCOPE_AGENT);
}

__device__ __forceinline__ void flush_seg(float* __restrict__ sum,
                                          float* __restrict__ sumsq,
                                          float* __restrict__ counts,
                                          int segId, const float4& a,
                                          const float4& q, int cnt, int lane) {
  if (segId < 0 || cnt == 0) return;
  float* sp = sum   + segId * DIM + lane * 4;
  float* qp = sumsq + segId * DIM + lane * 4;
  atom_add_f32(sp + 0, a.x); atom_add_f32(sp + 1, a.y);
  atom_add_f32(sp + 2, a.z); atom_add_f32(sp + 3, a.w);
  atom_add_f32(qp + 0, q.x); atom_add_f32(qp + 1, q.y);
  atom_add_f32(qp + 2, q.z); atom_add_f32(qp + 3, q.w);
  if (lane == 0) atom_add_f32(counts + segId, (float)cnt);
}

// ---------------------------------------------------------------------------
// Kernel 0: zero the accumulator region of the workspace
// ---------------------------------------------------------------------------
__global__ void gn_zero(float* __restrict__ p, int n) {
  int i = blockIdx.x * blockDim.x + threadIdx.x;
  if (i < n) p[i] = 0.0f;
}

// ---------------------------------------------------------------------------
// Kernel 1: per-segment sum(x), sum(x^2), counts.
// Each wave streams RPW contiguous rows through a double-buffered async-LDS
// pipeline, accumulates runs (segment ids are sorted) in registers, flushes
// with f32 atomics at segment boundaries only.
// ---------------------------------------------------------------------------
__global__ void __launch_bounds__(256) gn_pass1(
    const float* __restrict__ x, const int* __restrict__ seg,
    float* __restrict__ sum, float* __restrict__ sumsq,
    float* __restrict__ counts) {
  // 8 waves * 2 buffers * 8 rows * 32 lanes * 16B = 64KB
  __shared__ float4 lbuf[WAVES * 2 * TILE_ROWS * 32];

  const int lane = threadIdx.x & 31;
  const int wave = threadIdx.x >> 5;
  const int rowBase = (blockIdx.x * WAVES + wave) * RPW;
  if (rowBase >= N_ROWS) return;

  float4* buf[2] = { &lbuf[(wave * 2 + 0) * TILE_ROWS * 32],
                     &lbuf[(wave * 2 + 1) * TILE_ROWS * 32] };

  // Prologue: prefetch tile 0
  {
    const float* g = x + (size_t)rowBase * DIM + lane * 4;
#pragma unroll
    for (int r = 0; r < TILE_ROWS; ++r)
      async_load_row_b128(g + (size_t)r * DIM, buf[0] + r * 32 + lane);
  }

  int   runSeg = -1, runCnt = 0;
  float4 acc  = {0.f, 0.f, 0.f, 0.f};
  float4 accq = {0.f, 0.f, 0.f, 0.f};
  int   myseg = 0;

  for (int t = 0; t < TILES; ++t) {
    // Prefetch tile t+1 into the other buffer
    if (t + 1 < TILES) {
      float4* nb = buf[(t + 1) & 1];
      const float* g = x + (size_t)(rowBase + (t + 1) * TILE_ROWS) * DIM + lane * 4;
#pragma unroll
      for (int r = 0; r < TILE_ROWS; ++r)
        async_load_row_b128(g + (size_t)r * DIM, nb + r * 32 + lane);
    }
    // Refresh 32-row strip of segment ids (lane-resident, broadcast by shfl)
    if ((t & 3) == 0) myseg = seg[rowBase + (t >> 2) * 32 + lane];

    if (t + 1 < TILES) wait_async_le8(); else wait_async_le0();

    const float4* cb = buf[t & 1];
#pragma unroll
    for (int r = 0; r < TILE_ROWS; ++r) {
      const int lrow = t * TILE_ROWS + r;
      const int s = __shfl(myseg, lrow & 31, 32);   // wave-uniform
      if (s != runSeg) {
        flush_seg(sum, sumsq, counts, runSeg, acc, accq, runCnt, lane);
        runSeg = s; runCnt = 0;
        acc  = {0.f, 0.f, 0.f, 0.f};
        accq = {0.f, 0.f, 0.f, 0.f};
      }
      const float4 v = cb[r * 32 + lane];
      acc.x += v.x; acc.y += v.y; acc.z += v.z; acc.w += v.w;
      accq.x = fmaf(v.x, v.x, accq.x);
      accq.y = fmaf(v.y, v.y, accq.y);
      accq.z = fmaf(v.z, v.z, accq.z);
      accq.w = fmaf(v.w, v.w, accq.w);
      ++runCnt;
    }
  }
  flush_seg(sum, sumsq, counts, runSeg, acc, accq, runCnt, lane);
}

// ---------------------------------------------------------------------------
// Kernel 2: fold sums into per-(segment,feature) affine coeffs a, c.
// var = E[x^2] - (2s - s^2) * m^2 ; a = w/sqrt(var+eps) ; c = b - a*m*s
// ---------------------------------------------------------------------------
__global__ void __launch_bounds__(256) gn_stats(
    const float* __restrict__ sum, const float* __restrict__ sumsq,
    const float* __restrict__ counts, const float* __restrict__ w,
    const float* __restrict__ b, const float* __restrict__ ms,
    float* __restrict__ A, float* __restrict__ C) {
  const int idx = blockIdx.x * blockDim.x + threadIdx.x;   // < NSEG*DIM
  const int d = idx & (DIM - 1);
  const int g = idx >> 7;
  const float cnt = fmaxf(counts[g], 1.0f);
  const float m   = sum[idx]   / cnt;
  const float ex2 = sumsq[idx] / cnt;
  const float s   = ms[d];
  float var = ex2 - (2.0f - s) * s * m * m;
  var = fmaxf(var, 0.0f);
  const float a = w[d] / sqrtf(var + EPS_F);
  A[idx] = a;
  C[idx] = fmaf(-a * m, s, b[d]);                          // b - a*m*s
}

// ---------------------------------------------------------------------------
// Kernel 3: out = a[seg]*x + c[seg]  (streaming FMA; reload a,c only when the
// wave-uniform segment id changes — segments are sorted).
// x is dead after this read and out is write-once: use non-temporal hints so
// the 1 GB stream does not evict the hot A/C region from L2.
// ---------------------------------------------------------------------------
__global__ void __launch_bounds__(256) gn_pass3(
    const float* __restrict__ x, const int* __restrict__ seg,
    const float* __restrict__ A, const float* __restrict__ C,
    float* __restrict__ out) {
  const int lane = threadIdx.x & 31;
  const int wave = threadIdx.x >> 5;
  const int rowBase = (blockIdx.x * WAVES + wave) * RPW;
  if (rowBase >= N_ROWS) return;

  int curSeg = -1;
  v4f a4 = {0.f, 0.f, 0.f, 0.f}, c4 = {0.f, 0.f, 0.f, 0.f};

  for (int strip = 0; strip < RPW / 32; ++strip) {
    const int myseg = seg[rowBase + strip * 32 + lane];
#pragma unroll 4
    for (int r = 0; r < 32; ++r) {
      const int row = rowBase + strip * 32 + r;
      const int s = __shfl(myseg, r, 32);                  // wave-uniform
      if (s != curSeg) {
        curSeg = s;
        a4 = *(const v4f*)(A + (size_t)s * DIM + lane * 4);  // RT: keep in L2
        c4 = *(const v4f*)(C + (size_t)s * DIM + lane * 4);
      }
      const v4f v =
          __builtin_nontemporal_load((const v4f*)(x + (size_t)row * DIM + lane * 4));
      v4f o;
      o.x = fmaf(a4.x, v.x, c4.x);
      o.y = fmaf(a4.y, v.y, c4.y);
      o.z = fmaf(a4.z, v.z, c4.z);
      o.w = fmaf(a4.w, v.w, c4.w);
      __builtin_nontemporal_store(o, (v4f*)(out + (size_t)row * DIM + lane * 4));
    }
  }
}

// ---------------------------------------------------------------------------
// Host launcher
// ---------------------------------------------------------------------------
extern "C" void kernel_launch(void* const* d_in, const int* in_sizes, int n_in,
                              void* d_out, int out_size, void* d_ws, size_t ws_size,
                              hipStream_t stream) {
  const float* x   = (const float*)d_in[0];   // [N, D]
  const int*   seg = (const int*)d_in[1];     // [N], sorted
  // d_in[2] = num_segments scalar (compile-time NSEG=512)
  const float* w   = (const float*)d_in[3];   // [D]
  const float* b   = (const float*)d_in[4];   // [D]
  const float* ms  = (const float*)d_in[5];   // [D]
  float* out = (float*)d_out;

  float* ws     = (float*)d_ws;
  float* sum    = ws;                         // [NSEG*DIM]
  float* sumsq  = ws + NSEG * DIM;            // [NSEG*DIM]
  float* counts = ws + 2 * NSEG * DIM;        // [NSEG]
  float* A      = ws + 2 * NSEG * DIM + NSEG; // [NSEG*DIM]
  float* C      = A + NSEG * DIM;             // [NSEG*DIM]

  const int nzero = 2 * NSEG * DIM + NSEG;    // sum+sumsq+counts
  gn_zero<<<(nzero + 255) / 256, 256, 0, stream>>>(ws, nzero);

  gn_pass1<<<N_ROWS / ROWS_PER_BLOCK, 256, 0, stream>>>(x, seg, sum, sumsq, counts);

  gn_stats<<<(NSEG * DIM) / 256, 256, 0, stream>>>(sum, sumsq, counts, w, b, ms, A, C);

  gn_pass3<<<N_ROWS / ROWS_PER_BLOCK, 256, 0, stream>>>(x, seg, A, C, out);
}